// ReEig_st_50208167690569
// MI455X (gfx1250) — compile-verified
//
#include <hip/hip_runtime.h>
#include <math.h>

// ReEig for SPD matrices: batched Jacobi eigendecomposition + WMMA-f32 reconstruction.
// Input:  (4096, 64, 64) fp32 SPD.  Output: concat(out, u, S) flat fp32.

typedef __attribute__((ext_vector_type(2))) float v2f;
typedef __attribute__((ext_vector_type(8))) float v8f;

#define NDIM    64
#define LDP     66          // padded LDS row stride (even -> 8B aligned float2)
#define NSWEEPS 10
#define EPS_F   1e-4f

__global__ __launch_bounds__(256) void reeig_jacobi_kernel(
    const float* __restrict__ in, float* __restrict__ out, int nmat) {

  __shared__ float A_s[NDIM * LDP];
  __shared__ float V_s[NDIM * LDP];
  __shared__ float cP[32], sP[32];
  __shared__ int   pP[32], qP[32];
  __shared__ float ev[NDIM];     // raw diagonal eigenvalues
  __shared__ float sc[NDIM];     // clamped eigenvalues (unsorted order)
  __shared__ int   pos[NDIM];    // descending-sort rank of each eigenvalue

  const int tid = threadIdx.x;
  const int mat = blockIdx.x;

  // ---- load A into LDS, init V = I -------------------------------------
  const float* Am = in + (size_t)mat * (NDIM * NDIM);
  for (int it = 0; it < 16; ++it) {
    int idx = tid + it * 256;           // 4096 elements
    int r = idx >> 6, c = idx & 63;
    A_s[r * LDP + c] = Am[idx];
    V_s[r * LDP + c] = (r == c) ? 1.0f : 0.0f;
  }
  __syncthreads();

  // ---- cyclic parallel Jacobi (round-robin tournament pairing) ---------
  for (int sweep = 0; sweep < NSWEEPS; ++sweep) {
    for (int rr = 0; rr < 63; ++rr) {
      // 32 disjoint (p,q) pairs this round; lanes 0..31 compute rotations
      if (tid < 32) {
        int p, q;
        if (tid == 0) { p = 63; q = rr % 63; }
        else          { p = (rr + tid) % 63; q = (rr - tid + 63) % 63; }
        float app = A_s[p * LDP + p];
        float aqq = A_s[q * LDP + q];
        float apq = A_s[p * LDP + q];
        float c = 1.0f, s = 0.0f;
        if (fabsf(apq) > 1e-12f) {
          float tau = (aqq - app) / (2.0f * apq);
          float t   = copysignf(1.0f, tau) / (fabsf(tau) + sqrtf(1.0f + tau * tau));
          c = 1.0f / sqrtf(1.0f + t * t);
          s = t * c;
        }
        cP[tid] = c; sP[tid] = s; pP[tid] = p; qP[tid] = q;
      }
      __syncthreads();

      // rows p,q of A:  A <- J^T A     (pairs disjoint -> race-free)
      for (int it = 0; it < 8; ++it) {
        int idx = tid + it * 256;        // 32 pairs x 64 cols
        int pr = idx >> 6, k = idx & 63;
        int p = pP[pr], q = qP[pr];
        float c = cP[pr], s = sP[pr];
        float ap = A_s[p * LDP + k], aq = A_s[q * LDP + k];
        A_s[p * LDP + k] = c * ap - s * aq;
        A_s[q * LDP + k] = s * ap + c * aq;
      }
      __syncthreads();

      // cols p,q of A:  A <- A J ; and V <- V J
      for (int it = 0; it < 8; ++it) {
        int idx = tid + it * 256;
        int pr = idx >> 6, k = idx & 63;
        int p = pP[pr], q = qP[pr];
        float c = cP[pr], s = sP[pr];
        float akp = A_s[k * LDP + p], akq = A_s[k * LDP + q];
        A_s[k * LDP + p] = c * akp - s * akq;
        A_s[k * LDP + q] = s * akp + c * akq;
        float vkp = V_s[k * LDP + p], vkq = V_s[k * LDP + q];
        V_s[k * LDP + p] = c * vkp - s * vkq;
        V_s[k * LDP + q] = s * vkp + c * vkq;
      }
      __syncthreads();
    }
  }

  // ---- eigenvalues: clamp + descending sort rank -----------------------
  float* uR = out + (size_t)nmat * (NDIM * NDIM);
  float* sR = out + (size_t)nmat * (NDIM * NDIM) * 2;

  if (tid < 64) ev[tid] = A_s[tid * LDP + tid];
  __syncthreads();
  if (tid < 64) {
    float lam = ev[tid];
    int rnk = 0;
    for (int i = 0; i < 64; ++i) {
      float li = ev[i];
      rnk += (li > lam) || (li == lam && i < tid);
    }
    pos[tid] = rnk;
    sc[tid]  = fmaxf(lam, EPS_F);
    sR[(size_t)mat * 64 + rnk] = fmaxf(lam, EPS_F);
  }
  __syncthreads();

  // ---- write U with columns permuted to descending-eigenvalue order ----
  float* um = uR + (size_t)mat * (NDIM * NDIM);
  for (int it = 0; it < 16; ++it) {
    int idx = tid + it * 256;
    int r = idx >> 6, c = idx & 63;
    um[r * 64 + pos[c]] = V_s[r * LDP + c];
  }

  // ---- out = (V * diag(sc)) @ V^T via V_WMMA_F32_16X16X4_F32 -----------
  // (permutation-invariant, so use unsorted V / sc directly)
  const int wave = tid >> 5;          // 8 wave32s
  const int lane = tid & 31;
  const int lx = lane & 15;           // lane within half
  const int hx = lane >> 4;           // lane half select
  float* om = out + (size_t)mat * (NDIM * NDIM);

  for (int t = wave; t < 16; t += 8) {          // 4x4 grid of 16x16 tiles
    int mb = (t >> 2) * 16, nb = (t & 3) * 16;
    v8f acc = {};
    for (int k0 = 0; k0 < 64; k0 += 4) {
      int ka = k0 + 2 * hx;
      // A fragment: rows of W = V*diag(sc); ISA layout: lanes0-15 K=k0,k0+1 ; lanes16-31 K=k0+2,k0+3
      float2 av = *(const float2*)&V_s[(mb + lx) * LDP + ka];
      // B fragment: (V^T)[k][n] = V[n][k]; mirrored layout
      float2 bv = *(const float2*)&V_s[(nb + lx) * LDP + ka];
      v2f a; a.x = av.x * sc[ka]; a.y = av.y * sc[ka + 1];
      v2f b; b.x = bv.x;          b.y = bv.y;
      acc = __builtin_amdgcn_wmma_f32_16x16x4_f32(
          /*neg_a=*/false, a, /*neg_b=*/false, b,
          /*c_mod=*/(short)0, acc, /*reuse_a=*/false, /*reuse_b=*/false);
    }
    // D layout: lane holds out[mb + r + 8*hx][nb + lx] in acc[r]
    for (int r = 0; r < 8; ++r)
      om[(mb + r + 8 * hx) * 64 + (nb + lx)] = acc[r];
  }
}

extern "C" void kernel_launch(void* const* d_in, const int* in_sizes, int n_in,
                              void* d_out, int out_size, void* d_ws, size_t ws_size,
                              hipStream_t stream) {
  (void)n_in; (void)out_size; (void)d_ws; (void)ws_size;
  const float* in = (const float*)d_in[0];
  float* out = (float*)d_out;
  int nmat = in_sizes[0] / (NDIM * NDIM);   // 4096
  reeig_jacobi_kernel<<<dim3(nmat), dim3(256), 0, stream>>>(in, out, nmat);
}